// NeRFLoss_48988396978337
// MI455X (gfx1250) — compile-verified
//
#include <hip/hip_runtime.h>
#include <math.h>
#include <stdint.h>

#define R_RAYS 65536
#define S_SAMP 128
#define N_SAMP (R_RAYS * S_SAMP)

typedef float v4f __attribute__((ext_vector_type(4)));

// out layout: [0]=d_rgb, [1..R]=d_op, [R+1]=d_eik, [R+2..R+1+R]=d_dist
__global__ void nerf_init_kernel(float* __restrict__ out) {
    if (threadIdx.x == 0) {
        out[0] = 0.0f;
        out[R_RAYS + 1] = 0.0f;
    }
}

// Wave32 reduction helper: full 32-lane sum (result valid in lane 0 only).
__device__ __forceinline__ float wave_reduce_sum(float v) {
#pragma unroll
    for (int off = 16; off > 0; off >>= 1) v += __shfl_down(v, off, 32);
    return v;  // valid in lane 0
}

// Block (256 thread) reduction: wave shuffles + 8-entry LDS stage, then atomicAdd by thread 0.
__device__ __forceinline__ void block_reduce_atomic(float v, float scale, float* dst) {
    __shared__ float red[8];
    const int lane = threadIdx.x & 31;
    const int wave = threadIdx.x >> 5;
    v = wave_reduce_sum(v);
    if (lane == 0) red[wave] = v;
    __syncthreads();
    if (wave == 0) {
        float s = (lane < 8) ? red[lane] : 0.0f;
        s = wave_reduce_sum(s);
        if (lane == 0) atomicAdd(dst, s * scale);
    }
}

// One thread per ray: smooth-L1 over 3 rgb components (block-reduced) + opacity entropy.
__global__ __launch_bounds__(256) void nerf_rgb_op_kernel(const float* __restrict__ rgb,
                                                          const float* __restrict__ tgt,
                                                          const float* __restrict__ opac,
                                                          float* __restrict__ out) {
    const int i = blockIdx.x * 256 + threadIdx.x;
    float s = 0.0f;
#pragma unroll
    for (int c = 0; c < 3; ++c) {
        float d = fabsf(rgb[3 * i + c] - tgt[3 * i + c]);
        s += (d < 1.0f) ? 0.5f * d * d : d - 0.5f;
    }
    float o = opac[i];
    out[1 + i] = 0.05f * (-o * logf(o));   // L_OP * C_OP * entropy

    block_reduce_atomic(s, 1.0f / (3.0f * (float)R_RAYS), &out[0]);  // C_RGB=1
}

// 4 grad rows per thread via three 16B loads; mean((||g||-1)^2)*0.1.
// grad (96 MiB, read-once-per-pass) is loaded NON-TEMPORAL so it streams from HBM
// without evicting the L2-resident ws/ts/deltas working set between graph replays.
__global__ __launch_bounds__(256) void nerf_eik_kernel(const float* __restrict__ grad,
                                                       float* __restrict__ out) {
    const int t = blockIdx.x * 256 + threadIdx.x;          // 0 .. N/4-1
    const v4f* g4 = (const v4f*)grad + (size_t)t * 3;
    v4f a = __builtin_nontemporal_load(g4 + 0);
    v4f b = __builtin_nontemporal_load(g4 + 1);
    v4f c = __builtin_nontemporal_load(g4 + 2);
    float s = 0.0f;
    float n;
    n = sqrtf(a.x * a.x + a.y * a.y + a.z * a.z) - 1.0f; s += n * n;
    n = sqrtf(a.w * a.w + b.x * b.x + b.y * b.y) - 1.0f; s += n * n;
    n = sqrtf(b.z * b.z + b.w * b.w + c.x * c.x) - 1.0f; s += n * n;
    n = sqrtf(c.y * c.y + c.z * c.z + c.w * c.w) - 1.0f; s += n * n;

    block_reduce_atomic(s, 0.1f / (float)N_SAMP, &out[R_RAYS + 1]);
}

// One wave (32 lanes) per ray of 128 samples; 4 samples per lane.
// Streams staged into LDS with CDNA5 async global->LDS loads (ASYNCcnt path, default RT
// policy so these 96 MiB stay L2-resident across replays).
__global__ __launch_bounds__(256) void nerf_dist_kernel(const float* __restrict__ ws,
                                                        const float* __restrict__ deltas,
                                                        const float* __restrict__ ts,
                                                        const int* __restrict__ rays_a,
                                                        float* __restrict__ out) {
    __shared__ float ws_s[1024];
    __shared__ float ts_s[1024];
    __shared__ float ds_s[1024];

    const int tid  = threadIdx.x;
    const int lane = tid & 31;
    const int wave = tid >> 5;
    const int ray  = blockIdx.x * 8 + wave;
    const size_t gbase = (size_t)blockIdx.x * 1024 + (size_t)tid * 4;  // sample index

#if defined(__gfx1250__)
    {
        uint32_t lw = (uint32_t)(uintptr_t)&ws_s[tid * 4];
        uint32_t lt = (uint32_t)(uintptr_t)&ts_s[tid * 4];
        uint32_t ld = (uint32_t)(uintptr_t)&ds_s[tid * 4];
        const float* gw = ws + gbase;
        const float* gt = ts + gbase;
        const float* gd = deltas + gbase;
        asm volatile("global_load_async_to_lds_b128 %0, %1, off"
                     :: "v"(lw), "v"(gw) : "memory");
        asm volatile("global_load_async_to_lds_b128 %0, %1, off"
                     :: "v"(lt), "v"(gt) : "memory");
        asm volatile("global_load_async_to_lds_b128 %0, %1, off"
                     :: "v"(ld), "v"(gd) : "memory");
        // Each wave staged exactly its own ray's data -> per-wave wait, no barrier.
        asm volatile("s_wait_asynccnt 0" ::: "memory");
    }
#else
    ((v4f*)ws_s)[tid] = ((const v4f*)ws)[gbase / 4];
    ((v4f*)ts_s)[tid] = ((const v4f*)ts)[gbase / 4];
    ((v4f*)ds_s)[tid] = ((const v4f*)deltas)[gbase / 4];
#endif

    v4f w  = ((const v4f*)ws_s)[tid];
    v4f t4 = ((const v4f*)ts_s)[tid];
    v4f dl = ((const v4f*)ds_s)[tid];

    // per-lane sums of w and w*t over the lane's 4 contiguous samples
    float lw_sum  = w.x + w.y + w.z + w.w;
    float lwt_sum = w.x * t4.x + w.y * t4.y + w.z * t4.z + w.w * t4.w;

    // inclusive wave scan (wave32)
    float iw = lw_sum, iwt = lwt_sum;
#pragma unroll
    for (int off = 1; off < 32; off <<= 1) {
        float aw  = __shfl_up(iw, off, 32);
        float awt = __shfl_up(iwt, off, 32);
        if (lane >= off) { iw += aw; iwt += awt; }
    }
    float ew  = iw - lw_sum;    // exclusive prefix of sum(w) at this lane's first sample
    float ewt = iwt - lwt_sum;  // exclusive prefix of sum(w*t)

    float wl[4] = {w.x, w.y, w.z, w.w};
    float tl[4] = {t4.x, t4.y, t4.z, t4.w};
    float el[4] = {dl.x, dl.y, dl.z, dl.w};
    float acc = 0.0f;
#pragma unroll
    for (int j = 0; j < 4; ++j) {
        acc += 2.0f * wl[j] * (tl[j] * ew - ewt) + wl[j] * wl[j] * el[j] * (1.0f / 3.0f);
        ew  += wl[j];
        ewt += wl[j] * tl[j];
    }

    // wave reduction of acc
    acc = wave_reduce_sum(acc);

    if (lane == 0) {
        int target = rays_a[ray * 3];  // ray_idx (a permutation of 0..R-1)
        out[R_RAYS + 2 + target] = 0.001f * acc;  // L_DIST
    }
}

extern "C" void kernel_launch(void* const* d_in, const int* in_sizes, int n_in,
                              void* d_out, int out_size, void* d_ws, size_t ws_size,
                              hipStream_t stream) {
    (void)in_sizes; (void)n_in; (void)out_size; (void)d_ws; (void)ws_size;
    const float* rgb    = (const float*)d_in[0];
    const float* tgt    = (const float*)d_in[1];
    const float* opac   = (const float*)d_in[2];
    const float* grad   = (const float*)d_in[3];
    const float* ws     = (const float*)d_in[4];
    const float* deltas = (const float*)d_in[5];
    const float* ts     = (const float*)d_in[6];
    const int*   rays_a = (const int*)d_in[7];
    float* out = (float*)d_out;

    nerf_init_kernel<<<1, 32, 0, stream>>>(out);
    nerf_rgb_op_kernel<<<R_RAYS / 256, 256, 0, stream>>>(rgb, tgt, opac, out);
    nerf_eik_kernel<<<N_SAMP / 4 / 256, 256, 0, stream>>>(grad, out);
    nerf_dist_kernel<<<R_RAYS / 8, 256, 0, stream>>>(ws, deltas, ts, rays_a, out);
}